// RNN_36910948942510
// MI455X (gfx1250) — compile-verified
//
#include <hip/hip_runtime.h>

typedef float v2f __attribute__((ext_vector_type(2)));
typedef float v8f __attribute__((ext_vector_type(8)));

#define S_LEN 2048
#define I_DIM 128
#define H_DIM 512
#define O_DIM 64
#define B_DIM 32
#define M_ROWS (B_DIM * S_LEN)   // 65536

// ---------------------------------------------------------------------------
// Kernel A: xs[row] = x[row, 0:128] . input_receptive   (row = b*S + t)
// One wave per row, float4 loads, shfl-xor reduction.
// ---------------------------------------------------------------------------
__global__ __launch_bounds__(256) void xs_proj_kernel(
    const float* __restrict__ x, const float* __restrict__ inp_rec,
    float* __restrict__ xs_arr)
{
    const int lane = threadIdx.x & 31;
    const int row  = blockIdx.x * 8 + (threadIdx.x >> 5);
    const float4 xv = ((const float4*)(x + (size_t)row * I_DIM))[lane];
    const float4 iv = ((const float4*)inp_rec)[lane];
    float s = xv.x * iv.x + xv.y * iv.y + xv.z * iv.z + xv.w * iv.w;
#pragma unroll
    for (int m = 16; m >= 1; m >>= 1) s += __shfl_xor(s, m, 32);
    if (lane == 0) xs_arr[row] = s;
}

// ---------------------------------------------------------------------------
// Kernel B: scalar recursion
//   u_{t+1} = 0.9 u_t + 0.1 * sum_j rr_j * relu(u_t*rp_j + xs_t*ip_j)
// One wave per batch element; weights register-resident (16 per lane);
// in-wave shfl reduction only (no barriers on the 2048-long critical chain).
// Stores u_t (pre-update) for row (b,t).
// ---------------------------------------------------------------------------
__global__ __launch_bounds__(32) void u_scan_kernel(
    const float* __restrict__ xs_arr,
    const float* __restrict__ rec_rec,   // rr (H)
    const float* __restrict__ rec_proj,  // rp (H)
    const float* __restrict__ inp_proj,  // ip (H)
    float* __restrict__ u_arr)
{
    const int b    = blockIdx.x;
    const int lane = threadIdx.x;

    float rr[16], rp[16], ip[16];
#pragma unroll
    for (int c = 0; c < 16; ++c) {
        const int j = c * 32 + lane;
        rr[c] = rec_rec[j];
        rp[c] = rec_proj[j];
        ip[c] = inp_proj[j];
    }

    const float* __restrict__ xsb = xs_arr + b * S_LEN;
    float* __restrict__ ub        = u_arr  + b * S_LEN;

    float u = 0.0f;
    for (int t = 0; t < S_LEN; ++t) {
        if (lane == 0) ub[t] = u;      // row t of the GEMM uses pre-update u_t
        const float xs = xsb[t];
        float s0 = 0.f, s1 = 0.f, s2 = 0.f, s3 = 0.f;
#pragma unroll
        for (int c = 0; c < 16; c += 4) {
            s0 = fmaf(rr[c + 0], fmaxf(0.f, fmaf(u, rp[c + 0], xs * ip[c + 0])), s0);
            s1 = fmaf(rr[c + 1], fmaxf(0.f, fmaf(u, rp[c + 1], xs * ip[c + 1])), s1);
            s2 = fmaf(rr[c + 2], fmaxf(0.f, fmaf(u, rp[c + 2], xs * ip[c + 2])), s2);
            s3 = fmaf(rr[c + 3], fmaxf(0.f, fmaf(u, rp[c + 3], xs * ip[c + 3])), s3);
        }
        float s = (s0 + s1) + (s2 + s3);
#pragma unroll
        for (int m = 16; m >= 1; m >>= 1) s += __shfl_xor(s, m, 32);
        u = fmaf(0.9f, u, 0.1f * s);
    }
}

// ---------------------------------------------------------------------------
// Kernel C: g[m, o] = sum_j W[o,j] * relu(u[m]*rp[j] + xs[m]*ip[j])
// fp32 WMMA GEMM, M=65536 N=64 K=512.  A generated on the fly from (u,xs).
// 8 waves/block; each wave owns TWO 16-row M tiles and all 4 N tiles, so
// every B fragment (ds b64) feeds 2 WMMAs and each pq vector (ds b128) feeds 8.
// W staged in LDS as k-pair float2; (rp,ip) pairs in LDS.
//
// WMMA f32 16x16x4 lane layout (ISA 7.12.2):
//   A: lanes 0-15 -> M=lane, K=k0,k0+1 in a[0],a[1]; lanes 16-31 -> K=k0+2,k0+3
//   B: lanes 0-15 -> N=lane, K=k0,k0+1 in b[0],b[1]; lanes 16-31 -> K=k0+2,k0+3
//   C/D: vgpr v -> (M = v + 8*(lane>=16), N = lane&15)
// ---------------------------------------------------------------------------
__global__ __launch_bounds__(256) void wmma_readout_kernel(
    const float* __restrict__ u_arr, const float* __restrict__ xs_arr,
    const float* __restrict__ rec_proj, const float* __restrict__ inp_proj,
    const float* __restrict__ readout_w, float* __restrict__ g)
{
    __shared__ __align__(16) float2 lds_w[64 * 64];  // [kpair within chunk][n] 32KB
    __shared__ __align__(16) float2 lds_pq[H_DIM];   // (rp[k], ip[k])          4KB

    const int tid  = threadIdx.x;
    const int wave = tid >> 5;
    const int lane = tid & 31;
    const int lm   = lane & 15;
    const int hi   = lane >> 4;        // 0: K=k0,k0+1  1: K=k0+2,k0+3
    const int khalf = hi << 1;

    // stage (rp, ip) pairs once
    for (int k = tid; k < H_DIM; k += 256) {
        float2 pq;
        pq.x = rec_proj[k];
        pq.y = inp_proj[k];
        lds_pq[k] = pq;
    }

    // two M tiles per wave
    const int mbase0 = (blockIdx.x * 16 + wave * 2) * 16;
    const int mbase1 = mbase0 + 16;
    const float um0  = u_arr[mbase0 + lm];
    const float xs0  = xs_arr[mbase0 + lm];
    const float um1  = u_arr[mbase1 + lm];
    const float xs1  = xs_arr[mbase1 + lm];

    v8f acc0[4] = {};
    v8f acc1[4] = {};

    const float2* __restrict__ wsrc = (const float2*)readout_w; // pair (n, k/2) at n*256 + k/2

    for (int kc = 0; kc < 4; ++kc) {                 // K chunks of 128
        __syncthreads();
        for (int e = tid; e < 64 * 64; e += 256) {   // cooperative W chunk load
            const int n  = e >> 6;
            const int k2 = e & 63;                   // k-pair within chunk
            lds_w[k2 * 64 + n] = wsrc[n * 256 + kc * 64 + k2];
        }
        __syncthreads();

#pragma unroll 2
        for (int k0 = 0; k0 < 128; k0 += 4) {
            const int kk = k0 + khalf;               // this lane's first K (local)
            // A fragments: relu(u*rp + xs*ip) for K = kk, kk+1, both M tiles
            const float4 pq = *(const float4*)(lds_pq + (kc * 128 + kk));
            v2f a0, a1;
            a0[0] = fmaxf(0.f, fmaf(um0, pq.x, xs0 * pq.y));
            a0[1] = fmaxf(0.f, fmaf(um0, pq.z, xs0 * pq.w));
            a1[0] = fmaxf(0.f, fmaf(um1, pq.x, xs1 * pq.y));
            a1[1] = fmaxf(0.f, fmaf(um1, pq.z, xs1 * pq.w));

            const float2* __restrict__ brow = lds_w + (size_t)((kk >> 1) * 64) + lm;
#pragma unroll
            for (int nt = 0; nt < 4; ++nt) {
                const float2 bv = brow[nt * 16];
                v2f bf;
                bf[0] = bv.x;
                bf[1] = bv.y;
                acc0[nt] = __builtin_amdgcn_wmma_f32_16x16x4_f32(
                    false, a0, false, bf, (short)0, acc0[nt], false, false);
                acc1[nt] = __builtin_amdgcn_wmma_f32_16x16x4_f32(
                    false, a1, false, bf, (short)0, acc1[nt], false, false);
            }
        }
    }

    // write back: D vgpr v -> row mbase + v + 8*hi, col nt*16 + lm
    const int mrow0 = mbase0 + (hi << 3);
    const int mrow1 = mbase1 + (hi << 3);
#pragma unroll
    for (int v = 0; v < 8; ++v) {
        float* __restrict__ g0 = g + (size_t)(mrow0 + v) * O_DIM + lm;
        float* __restrict__ g1 = g + (size_t)(mrow1 + v) * O_DIM + lm;
#pragma unroll
        for (int nt = 0; nt < 4; ++nt) {
            g0[nt * 16] = acc0[nt][v];
            g1[nt * 16] = acc1[nt][v];
        }
    }
}

// ---------------------------------------------------------------------------
// Kernel D: exponential filter along t: y_t = 0.9 y_{t-1} + 0.1 g_t ; out = y + bias
// One thread per (b, o) chain; coalesced across o.
// ---------------------------------------------------------------------------
__global__ __launch_bounds__(64) void filter_kernel(
    const float* __restrict__ g, const float* __restrict__ bias,
    float* __restrict__ out)
{
    const int b = blockIdx.x;
    const int o = threadIdx.x;
    const float bo = bias[o];
    const float* __restrict__ gb = g   + (size_t)b * S_LEN * O_DIM + o;
    float* __restrict__ ob       = out + (size_t)b * S_LEN * O_DIM + o;
    float y = 0.0f;
#pragma unroll 8
    for (int t = 0; t < S_LEN; ++t) {
        y = fmaf(0.9f, y, 0.1f * gb[(size_t)t * O_DIM]);
        ob[(size_t)t * O_DIM] = y + bo;
    }
}

// ---------------------------------------------------------------------------
extern "C" void kernel_launch(void* const* d_in, const int* in_sizes, int n_in,
                              void* d_out, int out_size, void* d_ws, size_t ws_size,
                              hipStream_t stream) {
    (void)in_sizes; (void)n_in; (void)out_size; (void)ws_size;
    const float* x        = (const float*)d_in[0];  // (B,S,I)
    const float* inp_proj = (const float*)d_in[1];  // (H,1) -> ip[j]
    const float* inp_rec  = (const float*)d_in[2];  // (I,1)
    const float* rec_proj = (const float*)d_in[3];  // (H,1) -> rp[j]
    const float* rec_rec  = (const float*)d_in[4];  // (H,1) -> rr[j]
    const float* w        = (const float*)d_in[5];  // (O,H)
    const float* bias     = (const float*)d_in[6];  // (O)
    float* out            = (float*)d_out;          // (B,S,O)

    float* xs_arr = (float*)d_ws;                   // 65536 f32
    float* u_arr  = xs_arr + M_ROWS;                // 65536 f32
    float* g      = u_arr  + M_ROWS;                // 65536*64 f32 (16.8 MB)

    xs_proj_kernel<<<M_ROWS / 8, 256, 0, stream>>>(x, inp_rec, xs_arr);
    u_scan_kernel<<<B_DIM, 32, 0, stream>>>(xs_arr, rec_rec, rec_proj, inp_proj, u_arr);
    wmma_readout_kernel<<<M_ROWS / 256, 256, 0, stream>>>(
        u_arr, xs_arr, rec_proj, inp_proj, w, g);
    filter_kernel<<<B_DIM, O_DIM, 0, stream>>>(g, bias, out);
}